// NeuroTransNet_79869211836738
// MI455X (gfx1250) — compile-verified
//
#include <hip/hip_runtime.h>
#include <hip/hip_bf16.h>

// ================= model dimensions =================
#define BATCH 64
#define CHN   22
#define TLEN  1865
#define EMB   32
#define NHEAD 8
#define DKH   4
#define NOUTW 122          // (1865-50)/15 + 1
#define PK    50
#define PS    15
#define NCLS  4

#define KTAU  68           // 65 taps padded to 68 (mult of 4): 17 k-steps per channel
#define KCDIM (CHN*KTAU)   // 1496
#define KCSTP (KTAU/4)     // 17

#define TT    256          // time tile per front-conv block
#define XWIN  (TT+KTAU)    // 324 staged samples per channel
#define XSTR  328          // LDS row stride for staged x

// encoder LDS layout (floats), 128-row padded matrices, stride 33
#define MSTR   33
#define MROWS  128
#define MATF   (MROWS*MSTR)        // 4224
#define OFF_XS 0
#define OFF_HS (MATF)              // 4224
#define OFF_Q  (2*MATF)            // 8448
#define OFF_K  (3*MATF)
#define OFF_V  (4*MATF)
#define OFF_FF (2*MATF)            // union with Q/K/V (dead by FF time)
#define FFSTR  132
#define OFF_CP (2*MATF + MROWS*FFSTR)  // 25344
#define OFF_CA (OFF_CP+32)
#define OFF_CW (OFF_CA+16)
#define OFF_WB (OFF_CW+32)         // 25424: staged weights, 12288 floats
#define WB_Q   0
#define WB_K   1024
#define WB_V   2048
#define WB_O   3072
#define WB_F1  4096
#define WB_F2  8192
#define WB_TOT 12288
#define ENC_SMEM_FLOATS (OFF_WB + WB_TOT)   // 37712 floats = 150.8 KB
#define ENC_SMEM_BYTES  (ENC_SMEM_FLOATS*4)

typedef __attribute__((ext_vector_type(2))) float v2f;
typedef __attribute__((ext_vector_type(8))) float v8f;

static __device__ __forceinline__ v8f wmma_f32(v2f a, v2f b, v8f c) {
  // D(16x16,f32) = A(16x4,f32) x B(4x16,f32) + C
  return __builtin_amdgcn_wmma_f32_16x16x4_f32(false, a, false, b, (short)0, c, false, false);
}

// ---------- CDNA5 async LDS<->global (ASYNCcnt-tracked, VGPR-free copies) ----------
#define USE_ASYNC_LDS 1

static __device__ __forceinline__ unsigned lds_off(const void* p) {
  // LDS aperture keeps the in-wave LDS byte offset in addr[31:0]
  return (unsigned)(uintptr_t)p;
}
static __device__ __forceinline__ void async_ld_b32(void* lds, const float* g) {
#if USE_ASYNC_LDS
  asm volatile("global_load_async_to_lds_b32 %0, %1, off"
               :: "v"(lds_off(lds)), "v"((unsigned long long)(uintptr_t)g) : "memory");
#else
  *(volatile float*)lds = *g;
#endif
}
static __device__ __forceinline__ void async_st_b32(float* g, const void* lds) {
#if USE_ASYNC_LDS
  asm volatile("global_store_async_from_lds_b32 %0, %1, off"
               :: "v"((unsigned long long)(uintptr_t)g), "v"(lds_off(lds)) : "memory");
#else
  *g = *(volatile const float*)lds;
#endif
}
static __device__ __forceinline__ void wait_async0() {
#if USE_ASYNC_LDS
  asm volatile("s_wait_asynccnt 0x0" ::: "memory");
#endif
}

// ======================================================================
// Kernel 1: fold tconv->BN1->sconv->BN2 into Keff[32][22*68] and biasT[32]
// ======================================================================
__global__ __launch_bounds__(256)
void build_front_kernel(const float* tw0, const float* tw1, const float* tw2, const float* tw3,
                        const float* tb0, const float* tb1, const float* tb2, const float* tb3,
                        const float* bn1_g, const float* bn1_b,
                        const float* sconv_w, const float* sconv_b,
                        const float* bn2_g, const float* bn2_b,
                        float* __restrict__ Keff, float* __restrict__ biasT)
{
  const int   KER[4]  = {15, 25, 51, 65};
  const int   OFFS[4] = {32-7, 32-12, 32-25, 32-32};
  const float* tw[4] = {tw0, tw1, tw2, tw3};
  const float* tb[4] = {tb0, tb1, tb2, tb3};
  const float sbn = rsqrtf(1.0f + 1e-5f);              // BN uses rsqrt(1+eps), no stats
  const int tid = threadIdx.x;

  for (int idx = tid; idx < EMB*CHN*KTAU; idx += 256) {
    int o = idx / (CHN*KTAU);
    int rr = idx - o*(CHN*KTAU);
    int c = rr / KTAU;
    int tau = rr - c*KTAU;
    float acc = 0.f;
    if (tau < 65) {
      for (int i = 0; i < EMB; ++i) {
        int br = i >> 3, io = i & 7;
        int j = tau - OFFS[br];
        if (j >= 0 && j < KER[br]) {
          float s1 = bn1_g[i] * sbn;
          acc += sconv_w[(o*EMB + i)*CHN + c] * s1 * tw[br][io*KER[br] + j];
        }
      }
    }
    Keff[o*KCDIM + c*KTAU + tau] = acc * (bn2_g[o] * sbn);
  }

  if (tid < EMB) {
    int o = tid;
    float acc = sconv_b[o];
    for (int i = 0; i < EMB; ++i) {
      int br = i >> 3, io = i & 7;
      float add = tb[br][io] * (bn1_g[i] * sbn) + bn1_b[i];
      float sw = 0.f;
      for (int c = 0; c < CHN; ++c) sw += sconv_w[(o*EMB + i)*CHN + c];
      acc += sw * add;
    }
    biasT[o] = acc * (bn2_g[o] * sbn) + bn2_b[o];
  }
}

// ======================================================================
// Kernel 2: fused front conv (WMMA f32 GEMM over K=22*68) + bias + ELU
//   Each wave: one mtile, four ntiles (4 accumulators, 4x A reuse),
//   A-fragments double-buffered in registers (clause-loaded per channel).
// ======================================================================
__global__ __launch_bounds__(256)
void front_conv_kernel(const float* __restrict__ x, const float* __restrict__ Keff,
                       const float* __restrict__ biasT, float* __restrict__ h2)
{
  __shared__ float xs[CHN * XSTR];   // 28.9 KB
  const int tid = threadIdx.x;
  const int b = blockIdx.y;
  const int t0 = blockIdx.x * TT;

  // stage padded x window asynchronously; zero-fill OOB via LDS stores (overlaps)
  for (int e = tid; e < CHN*XWIN; e += 256) {
    int c = e / XWIN;
    int j = e - c*XWIN;
    int gt = t0 - 32 + j;
    float* dst = &xs[c*XSTR + j];
    if (gt >= 0 && gt < TLEN) async_ld_b32(dst, &x[((size_t)b*CHN + c)*TLEN + gt]);
    else                      *dst = 0.f;
  }
  wait_async0();
  __syncthreads();

  const int lane = tid & 31, wave = tid >> 5;
  const int n = lane & 15, half = lane >> 4;
  const int mt  = wave >> 2;            // waves 0-3: mtile 0, waves 4-7: mtile 1
  const int ntb = (wave & 3) * 4;       // four consecutive ntiles per wave
  const int o_row = mt*16 + n;

  v8f acc[4];
  #pragma unroll
  for (int jj = 0; jj < 4; ++jj) acc[jj] = (v8f){0.f,0.f,0.f,0.f,0.f,0.f,0.f,0.f};

  const float* ar = Keff + (size_t)o_row*KCDIM + 2*half;

  auto prefA = [&](v2f* dst, const float* src) {
    #pragma unroll
    for (int s = 0; s < KCSTP; ++s) dst[s] = *(const v2f*)(src + 4*s);
  };
  auto compute = [&](const v2f* areg, int c) {
    #pragma unroll
    for (int jj = 0; jj < 4; ++jj) {
      const float* bp = &xs[c*XSTR + (ntb + jj)*16 + n + 2*half];
      #pragma unroll
      for (int s = 0; s < KCSTP; ++s) {
        v2f bv; bv.x = bp[0]; bv.y = bp[1];  bp += 4;
        acc[jj] = wmma_f32(areg[s], bv, acc[jj]);
      }
    }
  };

  v2f a0[KCSTP], a1[KCSTP];
  prefA(a0, ar);
  for (int c = 0; c < CHN; c += 2) {           // CHN = 22, even
    prefA(a1, ar + KTAU);                      // next channel in flight
    compute(a0, c);
    if (c + 2 < CHN) prefA(a0, ar + 2*KTAU);
    compute(a1, c + 1);
    ar += 2*KTAU;
  }

  #pragma unroll
  for (int jj = 0; jj < 4; ++jj) {
    const int t = t0 + (ntb + jj)*16 + n;
    if (t < TLEN) {
      for (int r = 0; r < 8; ++r) {
        int o = mt*16 + r + 8*half;            // D row layout
        float v = acc[jj][r] + biasT[o];
        h2[((size_t)b*EMB + o)*TLEN + t] = (v > 0.f) ? v : expm1f(v);
      }
    }
  }
}

// ======================================================================
// Kernel 3: windowed mean / log-var pooling, one block per (b,o) row
// ======================================================================
__global__ __launch_bounds__(256)
void pool_kernel(const float* __restrict__ h2, float* __restrict__ x1, float* __restrict__ x2)
{
  __shared__ float row[TLEN + 7];
  const int bo = blockIdx.x;
  const int b = bo >> 5, o = bo & 31;
  const float* src = h2 + ((size_t)b*EMB + o)*TLEN;
  for (int e = threadIdx.x; e < TLEN; e += 256) async_ld_b32(&row[e], src + e);
  wait_async0();
  __syncthreads();
  for (int nw = threadIdx.x; nw < NOUTW; nw += 256) {
    int st = nw * PS;
    float s = 0.f, sq = 0.f;
    for (int j = 0; j < PK; ++j) { float v = row[st + j]; s += v; sq += v*v; }
    float mean = s * (1.f/PK);
    float var = (sq - s*mean) * (1.f/(PK-1));          // ddof=1
    var = fminf(fmaxf(var, 1e-6f), 1e6f);
    size_t oi = ((size_t)b*NOUTW + nw)*EMB + o;
    x1[oi] = mean;
    x2[oi] = logf(var);
  }
}

// ======================================================================
// Kernel 4: one full encoder layer, one block per (batch,stream)
// ======================================================================
struct EncW {
  const float *ln1_g, *ln1_b, *ln2_g, *ln2_b;
  const float *wq_w, *wq_b, *wk_w, *wk_b, *wv_w, *wv_b, *wo_w, *wo_b;
  const float *sa1_w, *sa1_b, *sa2_w, *sa2_b, *ca1_w, *ca1_b, *ca2_w, *ca2_b;
  const float *ff1_w, *ff1_b, *ff2_w, *ff2_b;
};

__global__ __launch_bounds__(256)
void encoder_kernel(float* __restrict__ x1, float* __restrict__ x2, EncW W)
{
  extern __shared__ float sm[];
  float* Xs = sm + OFF_XS;
  float* Hs = sm + OFF_HS;
  float* Qs = sm + OFF_Q;
  float* Ks = sm + OFF_K;
  float* Vs = sm + OFF_V;
  float* FF = sm + OFF_FF;
  float* CP = sm + OFF_CP;
  float* CA = sm + OFF_CA;
  float* CW = sm + OFF_CW;
  float* WB = sm + OFF_WB;

  const int tid = threadIdx.x;
  const int lane = tid & 31, wave = tid >> 5;
  const int bs = blockIdx.x;
  float* xg = ((bs & 1) ? x2 : x1) + (size_t)(bs >> 1) * (NOUTW*EMB);

  // async: token load + all six GEMM weight matrices into LDS, one wait
  for (int e = tid; e < NOUTW*EMB; e += 256)
    async_ld_b32(&Xs[(e>>5)*MSTR + (e&31)], xg + e);
  for (int e = tid; e < WB_TOT; e += 256) {
    const float* src;
    if      (e < 1024) src = W.wq_w + e;
    else if (e < 2048) src = W.wk_w + (e - 1024);
    else if (e < 3072) src = W.wv_w + (e - 2048);
    else if (e < 4096) src = W.wo_w + (e - 3072);
    else if (e < 8192) src = W.ff1_w + (e - 4096);
    else               src = W.ff2_w + (e - 8192);
    async_ld_b32(&WB[e], src);
  }
  for (int e = tid; e < (MROWS-NOUTW)*EMB; e += 256)
    Xs[(NOUTW + (e>>5))*MSTR + (e&31)] = 0.f;
  wait_async0();
  __syncthreads();

  auto layernorm = [&](const float* src, float* dst, const float* g, const float* bta) {
    if (tid < MROWS) {
      int t = tid;
      float s = 0.f;
      for (int f = 0; f < EMB; ++f) s += src[t*MSTR + f];
      float mean = s * (1.f/EMB), vs = 0.f;
      for (int f = 0; f < EMB; ++f) { float d = src[t*MSTR + f] - mean; vs += d*d; }
      float inv = rsqrtf(vs * (1.f/EMB) + 1e-5f);
      for (int f = 0; f < EMB; ++f)
        dst[t*MSTR + f] = (src[t*MSTR + f] - mean) * inv * g[f] + bta[f];
    }
  };

  // f32 WMMA GEMM, both operands in LDS; optional per-k A scale
  auto gemm = [&](const float* A, int lda, const float* Bl, int ldb,
                  int MT, int NT, int ksteps, const float* ascale, auto epi) {
    const int n = lane & 15, half = lane >> 4;
    for (int job = wave; job < MT*NT; job += 8) {
      const int mt = job / NT, nt = job - mt*NT;
      const float* ar = A + (mt*16 + n)*lda + 2*half;
      const float* br = Bl + (2*half)*ldb + nt*16 + n;
      v8f acc = {0.f,0.f,0.f,0.f,0.f,0.f,0.f,0.f};
      for (int s = 0; s < ksteps; ++s) {
        v2f a; a.x = ar[0]; a.y = ar[1];
        if (ascale) { int k = s*4 + 2*half; a.x *= ascale[k]; a.y *= ascale[k+1]; }
        v2f b; b.x = br[0]; b.y = br[ldb];
        acc = wmma_f32(a, b, acc);
        ar += 4; br += 4*(size_t)ldb;
      }
      for (int r = 0; r < 8; ++r) epi(mt*16 + r + 8*half, nt*16 + n, acc[r]);
    }
  };

  // ---- LN1 ----
  layernorm(Xs, Hs, W.ln1_g, W.ln1_b);
  __syncthreads();

  // ---- Q, K, Vflat (weights from LDS) ----
  gemm(Hs, MSTR, WB + WB_Q, EMB, 8, 2, 8, nullptr,
       [&](int row, int col, float v){ Qs[row*MSTR + col] = v + W.wq_b[col]; });
  gemm(Hs, MSTR, WB + WB_K, EMB, 8, 2, 8, nullptr,
       [&](int row, int col, float v){ Ks[row*MSTR + col] = v + W.wk_b[col]; });
  gemm(Hs, MSTR, WB + WB_V, EMB, 8, 2, 8, nullptr,
       [&](int row, int col, float v){ Vs[row*MSTR + col] = v + W.wv_b[col]; });
  __syncthreads();

  // ---- spatial gate: Vs *= sigmoid(relu(Vs@sa1+b)@sa2+b)  (per-token, scalar) ----
  if (tid < NOUTW) {
    const int t = tid;
    float a16[16];
    for (int j = 0; j < 16; ++j) {
      float s = W.sa1_b[j];
      for (int f = 0; f < EMB; ++f) s += Vs[t*MSTR + f] * W.sa1_w[f*16 + j];
      a16[j] = fmaxf(s, 0.f);
    }
    for (int f = 0; f < EMB; ++f) {
      float s = W.sa2_b[f];
      for (int j = 0; j < 16; ++j) s += a16[j] * W.sa2_w[j*EMB + f];
      Vs[t*MSTR + f] *= 1.f / (1.f + __expf(-s));
    }
  }
  // ---- channel gate prep: cp = mean over tokens of Hs (other threads) ----
  if (tid >= 128 && tid < 128 + EMB) {
    int f = tid - 128;
    float s = 0.f;
    for (int t = 0; t < NOUTW; ++t) s += Hs[t*MSTR + f];
    CP[f] = s * (1.f/NOUTW);
  }
  __syncthreads();
  if (tid < 16) {
    float s = W.ca1_b[tid];
    for (int f = 0; f < EMB; ++f) s += CP[f] * W.ca1_w[f*16 + tid];
    CA[tid] = fmaxf(s, 0.f);
  }
  __syncthreads();
  if (tid < EMB) {
    float s = W.ca2_b[tid];
    for (int j = 0; j < 16; ++j) s += CA[j] * W.ca2_w[j*EMB + tid];
    CW[tid] = 1.f / (1.f + __expf(-s));
  }
  __syncthreads();

  // ---- V += (Hs * cw) @ wv + bv ----
  gemm(Hs, MSTR, WB + WB_V, EMB, 8, 2, 8, CW,
       [&](int row, int col, float v){ Vs[row*MSTR + col] += v + W.wv_b[col]; });
  __syncthreads();

  // ---- attention: dk=4, online softmax per (head, query); output -> Hs ----
  for (int wi = tid; wi < NHEAD*NOUTW; wi += 256) {
    const int h = wi / NOUTW, q = wi - h*NOUTW;
    const float* qp = Qs + q*MSTR + h*DKH;
    const float q0 = qp[0], q1 = qp[1], q2 = qp[2], q3 = qp[3];
    float mx = -3.4e38f, l = 0.f, a0 = 0.f, a1 = 0.f, a2 = 0.f, a3 = 0.f;
    for (int k = 0; k < NOUTW; ++k) {
      const float* kp = Ks + k*MSTR + h*DKH;
      float s = (q0*kp[0] + q1*kp[1] + q2*kp[2] + q3*kp[3]) * 0.5f;  // /sqrt(4)
      float nm = fmaxf(mx, s);
      float c = __expf(mx - nm), p = __expf(s - nm);
      const float* vp = Vs + k*MSTR + h*DKH;
      l = l*c + p;
      a0 = a0*c + p*vp[0]; a1 = a1*c + p*vp[1];
      a2 = a2*c + p*vp[2]; a3 = a3*c + p*vp[3];
      mx = nm;
    }
    float inv = 1.f / l;
    float* op = Hs + q*MSTR + h*DKH;   // Hs reuse (LN1 output fully consumed)
    op[0] = a0*inv; op[1] = a1*inv; op[2] = a2*inv; op[3] = a3*inv;
  }
  __syncthreads();

  // ---- X += attn_out @ wo + bo ----
  gemm(Hs, MSTR, WB + WB_O, EMB, 8, 2, 8, nullptr,
       [&](int row, int col, float v){ Xs[row*MSTR + col] += v + W.wo_b[col]; });
  __syncthreads();

  // ---- LN2 ----
  layernorm(Xs, Hs, W.ln2_g, W.ln2_b);
  __syncthreads();

  // ---- FF1 + exact GELU ----
  gemm(Hs, MSTR, WB + WB_F1, 128, 8, 8, 8, nullptr,
       [&](int row, int col, float v){
         v += W.ff1_b[col];
         FF[row*FFSTR + col] = 0.5f * v * (1.f + erff(v * 0.70710678118654752f));
       });
  __syncthreads();

  // ---- X += FF @ ff2 + b2 ----
  gemm(FF, FFSTR, WB + WB_F2, EMB, 8, 2, 32, nullptr,
       [&](int row, int col, float v){ Xs[row*MSTR + col] += v + W.ff2_b[col]; });
  __syncthreads();

  // ---- async writeback (LDS -> global, no VGPR round trip) ----
  for (int e = tid; e < NOUTW*EMB; e += 256)
    async_st_b32(xg + e, &Xs[(e>>5)*MSTR + (e&31)]);
  wait_async0();
}

// ======================================================================
// Kernel 5: channel-expand + BN3 + ELU + classifier, one block per batch
// ======================================================================
__global__ __launch_bounds__(256)
void head_kernel(const float* __restrict__ x1, const float* __restrict__ x2,
                 const float* __restrict__ ce_w, const float* __restrict__ ce_b,
                 const float* __restrict__ bn3_g, const float* __restrict__ bn3_b,
                 const float* __restrict__ cls_w, const float* __restrict__ cls_b,
                 float* __restrict__ out)
{
  __shared__ float yv[64*EMB];
  __shared__ float red[256*NCLS];
  const int b = blockIdx.x, tid = threadIdx.x;
  const float* p1 = x1 + (size_t)b*NOUTW*EMB;
  const float* p2 = x2 + (size_t)b*NOUTW*EMB;
  const float sbn = rsqrtf(1.f + 1e-5f);

  for (int e = tid; e < 64*EMB; e += 256) {
    int o = e >> 5, w = e & 31;
    float acc = ce_b[o];
    const float* cw = ce_w + o*(NOUTW*2);
    for (int i = 0; i < NOUTW; ++i)
      acc += p1[i*EMB + w]*cw[i*2 + 0] + p2[i*EMB + w]*cw[i*2 + 1];
    acc = acc * (bn3_g[o]*sbn) + bn3_b[o];
    yv[e] = (acc > 0.f) ? acc : expm1f(acc);
  }
  __syncthreads();
  float p[NCLS] = {0.f, 0.f, 0.f, 0.f};
  for (int e = tid; e < 64*EMB; e += 256) {
    float v = yv[e];
    p[0] += v*cls_w[e*4+0]; p[1] += v*cls_w[e*4+1];
    p[2] += v*cls_w[e*4+2]; p[3] += v*cls_w[e*4+3];
  }
  for (int c = 0; c < NCLS; ++c) red[tid*NCLS + c] = p[c];
  __syncthreads();
  for (int st = 128; st > 0; st >>= 1) {
    if (tid < st)
      for (int c = 0; c < NCLS; ++c) red[tid*NCLS + c] += red[(tid+st)*NCLS + c];
    __syncthreads();
  }
  if (tid < NCLS) out[b*NCLS + tid] = red[tid] + cls_b[tid];
}

// ======================================================================
// launch
// ======================================================================
// d_in flattening assumed: insertion order of setup_inputs()/params dicts:
//  0:x 1-4:tconv_w[0..3] 5-8:tconv_b[0..3] 9:bn1_g 10:bn1_b 11:sconv_w 12:sconv_b
//  13:bn2_g 14:bn2_b 15:ce_w 16:ce_b 17:bn3_g 18:bn3_b 19:cls_w 20:cls_b
//  21+24*L: layer L leaves in declaration order
extern "C" void kernel_launch(void* const* d_in, const int* in_sizes, int n_in,
                              void* d_out, int out_size, void* d_ws, size_t ws_size,
                              hipStream_t stream)
{
  (void)in_sizes; (void)n_in; (void)out_size; (void)ws_size;
  #define F(i) ((const float*)d_in[(i)])

  float* ws = (float*)d_ws;
  float* Keff  = ws;                                  // 32*1496          = 47872
  float* biasT = ws + 47872;                          // 32
  float* h2    = ws + 47904;                          // 64*32*1865       = 3819520
  float* x1b   = ws + 47904 + 3818720;                // 64*122*32        = 249856
  float* x2b   = x1b + 249856;                        // (total ~16.7 MB workspace)

  build_front_kernel<<<1, 256, 0, stream>>>(
      F(1), F(2), F(3), F(4), F(5), F(6), F(7), F(8),
      F(9), F(10), F(11), F(12), F(13), F(14), Keff, biasT);

  dim3 fcgrid((TLEN + TT - 1)/TT, BATCH);
  front_conv_kernel<<<fcgrid, 256, 0, stream>>>(F(0), Keff, biasT, h2);

  pool_kernel<<<BATCH*EMB, 256, 0, stream>>>(h2, x1b, x2b);

  for (int L = 0; L < 4; ++L) {
    const int base = 21 + L*24;
    EncW W;
    W.ln1_g = F(base+0);  W.ln1_b = F(base+1);  W.ln2_g = F(base+2);  W.ln2_b = F(base+3);
    W.wq_w  = F(base+4);  W.wq_b  = F(base+5);  W.wk_w  = F(base+6);  W.wk_b  = F(base+7);
    W.wv_w  = F(base+8);  W.wv_b  = F(base+9);  W.wo_w  = F(base+10); W.wo_b  = F(base+11);
    W.sa1_w = F(base+12); W.sa1_b = F(base+13); W.sa2_w = F(base+14); W.sa2_b = F(base+15);
    W.ca1_w = F(base+16); W.ca1_b = F(base+17); W.ca2_w = F(base+18); W.ca2_b = F(base+19);
    W.ff1_w = F(base+20); W.ff1_b = F(base+21); W.ff2_w = F(base+22); W.ff2_b = F(base+23);
    encoder_kernel<<<BATCH*2, 256, ENC_SMEM_BYTES, stream>>>(x1b, x2b, W);
  }

  head_kernel<<<BATCH, 256, 0, stream>>>(x1b, x2b, F(15), F(16), F(17), F(18),
                                         F(19), F(20), (float*)d_out);
  #undef F
}